// JointLoss_17669495456297
// MI455X (gfx1250) — compile-verified
//
#include <hip/hip_runtime.h>
#include <hip/hip_bf16.h>
#include <math.h>

typedef __attribute__((ext_vector_type(2))) float v2f;
typedef __attribute__((ext_vector_type(8))) float v8f;

#define TAU_INV 10.0f     // 1 / 0.1
#define NROWS   4096
#define DIMS    256

// ---------------------------------------------------------------------------
// recon_loss partials: grid-stride float4 over 8M elements, per-block sum
// ---------------------------------------------------------------------------
__global__ void recon_kernel(const float* __restrict__ xr,
                             const float* __restrict__ xo,
                             float* __restrict__ partials, int n4) {
    __shared__ float red[256];
    const int tid = threadIdx.x;
    int idx = blockIdx.x * blockDim.x + tid;
    const int stride = gridDim.x * blockDim.x;
    const float4* a = (const float4*)xr;
    const float4* b = (const float4*)xo;
    float acc = 0.f;
    for (int i = idx; i < n4; i += stride) {
        float4 u = a[i], v = b[i];
        float d0 = u.x - v.x, d1 = u.y - v.y, d2 = u.z - v.z, d3 = u.w - v.w;
        acc += d0 * d0 + d1 * d1 + d2 * d2 + d3 * d3;
    }
    red[tid] = acc;
    __syncthreads();
    for (int s = 128; s > 0; s >>= 1) {
        if (tid < s) red[tid] += red[tid + s];
        __syncthreads();
    }
    if (tid == 0) partials[blockIdx.x] = red[0];
}

// ---------------------------------------------------------------------------
// Contrastive loss: one block per 16-row slab. sim = P * P^T / tau computed
// with V_WMMA_F32_16X16X4_F32; per-row pos/neg exp-sums accumulated without
// ever materializing the 4096x4096 sim matrix.
//
// WMMA fragment layout (CDNA5 ISA, 32-bit 16x16x4):
//   A (16x4):  lane L holds A[L&15][2*(L>>4) + {0,1}]  (v2f)
//   B (4x16):  lane L holds B[2*(L>>4) + {0,1}][L&15]  (v2f)  == P[col][k..]
//   C/D:       VGPR j, lane L = C[j + 8*(L>>4)][L&15]  (v8f)
// ---------------------------------------------------------------------------
__global__ void contrast_kernel(const float* __restrict__ P,
                                float* __restrict__ partials) {
    __shared__ float ldsA[16 * DIMS];   // 16 KB: this block's A rows
    __shared__ float ldsPos[8][16];
    __shared__ float ldsNeg[8][16];
    __shared__ float ldsRow[16];

    const int tid = threadIdx.x;
    const int rowBase = blockIdx.x * 16;

    // stage 16 contiguous rows of P into LDS (coalesced)
    for (int i = tid; i < 16 * DIMS; i += 256)
        ldsA[i] = P[rowBase * DIMS + i];
    __syncthreads();

    const int wave = tid >> 5;
    const int lane = tid & 31;
    const int half = lane >> 4;   // which K-pair of the K=4 chunk
    const int m    = lane & 15;   // A-row / B-col within tile

    float pos[8], neg[8];
    int rowGrp[8];                // group id of the row each C slot belongs to
#pragma unroll
    for (int j = 0; j < 8; ++j) {
        pos[j] = 0.f;
        neg[j] = 0.f;
        rowGrp[j] = (rowBase + j + 8 * half) >> 2;
    }

    const float* arow = ldsA + m * DIMS;

    // each wave owns column tiles wave, wave+8, ... (32 tiles per wave)
    for (int tile = wave; tile < 256; tile += 8) {
        const int col = tile * 16 + m;
        const float* brow = P + (size_t)col * DIMS;

        v8f c = {0.f, 0.f, 0.f, 0.f, 0.f, 0.f, 0.f, 0.f};
#pragma unroll 8
        for (int kc = 0; kc < 64; ++kc) {       // K = 256 in chunks of 4
            v2f a = *(const v2f*)(arow + 4 * kc + 2 * half);
            v2f b = *(const v2f*)(brow + 4 * kc + 2 * half);
            c = __builtin_amdgcn_wmma_f32_16x16x4_f32(
                    false, a, false, b, (short)0, c, false, false);
        }

        const int cg = col >> 2;                // column's group id
#pragma unroll
        for (int j = 0; j < 8; ++j) {
            const float sim = c[j] * TAU_INV;
            const float e = __expf(sim);
            const bool grp = (rowGrp[j] == cg);
            // branchless: pos_mask = group & (sim != 1); neg = !group
            pos[j] += (grp && (sim != 1.0f)) ? e : 0.0f;
            neg[j] += grp ? 0.0f : e;
        }
    }

    // reduce over the 16 lanes of each half (columns of the tiles)
#pragma unroll
    for (int j = 0; j < 8; ++j) {
        for (int off = 1; off < 16; off <<= 1) {
            pos[j] += __shfl_xor(pos[j], off, 32);
            neg[j] += __shfl_xor(neg[j], off, 32);
        }
    }
    if (m == 0) {
#pragma unroll
        for (int j = 0; j < 8; ++j) {
            ldsPos[wave][j + 8 * half] = pos[j];
            ldsNeg[wave][j + 8 * half] = neg[j];
        }
    }
    __syncthreads();

    // combine the 8 waves' partial sums, per-row -log(pos/(pos+neg))
    if (tid < 16) {
        float p = 0.f, ng = 0.f;
        for (int w = 0; w < 8; ++w) { p += ldsPos[w][tid]; ng += ldsNeg[w][tid]; }
        ldsRow[tid] = -__logf(p / (p + ng));
    }
    __syncthreads();
    if (tid == 0) {
        float s = 0.f;
        for (int r = 0; r < 16; ++r) s += ldsRow[r];
        partials[blockIdx.x] = s;
    }
}

// ---------------------------------------------------------------------------
// dist_loss partials: 6 intra-group pair diffs per (group, dim)
// ---------------------------------------------------------------------------
__global__ void dist_kernel(const float* __restrict__ P,
                            float* __restrict__ partials) {
    __shared__ float red[256];
    const int tid = threadIdx.x;
    const int gidx = blockIdx.x * 256 + tid;   // 0..65535
    float acc = 0.f;
#pragma unroll
    for (int rep = 0; rep < 4; ++rep) {
        const int e = gidx + rep * 65536;      // 0..262143 = (group, dim)
        const int g = e >> 8;
        const int d = e & 255;
        const float* base = P + (size_t)g * (4 * DIMS) + d;
        float a0 = base[0], a1 = base[DIMS], a2 = base[2 * DIMS], a3 = base[3 * DIMS];
        float d01 = a0 - a1, d02 = a0 - a2, d03 = a0 - a3;
        float d12 = a1 - a2, d13 = a1 - a3, d23 = a2 - a3;
        acc += d01 * d01 + d02 * d02 + d03 * d03 +
               d12 * d12 + d13 * d13 + d23 * d23;
    }
    red[tid] = acc;
    __syncthreads();
    for (int s = 128; s > 0; s >>= 1) {
        if (tid < s) red[tid] += red[tid + s];
        __syncthreads();
    }
    if (tid == 0) partials[blockIdx.x] = red[0];
}

// ---------------------------------------------------------------------------
// Deterministic final reduction: fixed-order sums of all partials
// ---------------------------------------------------------------------------
__global__ void final_kernel(const float* __restrict__ reconP, int nRecon,
                             const float* __restrict__ clossP, int nCloss,
                             const float* __restrict__ distP, int nDist,
                             float* __restrict__ out) {
    float sr = 0.f, sc = 0.f, sd = 0.f;
    for (int i = 0; i < nRecon; ++i) sr += reconP[i];
    for (int i = 0; i < nCloss; ++i) sc += clossP[i];
    for (int i = 0; i < nDist;  ++i) sd += distP[i];
    const float recon = sr / (4096.0f * 2048.0f);
    const float closs = sc / 4096.0f;
    const float dist  = sd / (1024.0f * 6.0f * 256.0f);
    out[0] = recon + closs + dist;   // loss
    out[1] = closs;
    out[2] = recon;
    out[3] = dist;
}

extern "C" void kernel_launch(void* const* d_in, const int* in_sizes, int n_in,
                              void* d_out, int out_size, void* d_ws, size_t ws_size,
                              hipStream_t stream) {
    const float* proj = (const float*)d_in[0];   // (4096, 256)
    const float* xrec = (const float*)d_in[1];   // (4096, 2048)
    const float* xorg = (const float*)d_in[2];   // (4096, 2048)

    float* ws     = (float*)d_ws;
    float* reconP = ws;            // 1024 floats
    float* clossP = ws + 1024;     // 256 floats
    float* distP  = ws + 1280;     // 256 floats

    recon_kernel   <<<1024, 256, 0, stream>>>(xrec, xorg, reconP, (NROWS * 2048) / 4);
    contrast_kernel<<< 256, 256, 0, stream>>>(proj, clossP);
    dist_kernel    <<< 256, 256, 0, stream>>>(proj, distP);
    final_kernel   <<<   1,   1, 0, stream>>>(reconP, 1024, clossP, 256, distP, 256,
                                              (float*)d_out);
}